// BBoxProposer_70411693851324
// MI455X (gfx1250) — compile-verified
//
#include <hip/hip_runtime.h>
#include <hip/hip_bf16.h>
#include <stdint.h>

// ---------------- problem constants ----------------
#define S_    32
#define A_    3
#define ATTRS 85
#define H_    52
#define W_    52
#define HW    (H_ * W_)          // 2704
#define NBOX  (S_ * A_ * HW)     // 259584
#define KTOP  4096
#define CAP   8192               // candidate cap (expected ~3.6K valid)
#define WORDS (KTOP / 32)        // 128 mask words per row
#define OBJ_THR 0.9f

typedef unsigned int       u32;
typedef unsigned long long u64;
typedef float v2f __attribute__((ext_vector_type(2)));
typedef float v8f __attribute__((ext_vector_type(8)));

// ---------------- workspace layout (bytes) ----------------
#define OFF_CNT    0
#define OFF_KEYS   64
#define OFF_SLOTS  (OFF_KEYS + CAP * 8)          // CAP * 6 floats
#define OFF_CAND   (OFF_SLOTS + CAP * 6 * 4)     // KTOP * 6 floats
#define OFF_X1     (OFF_CAND + KTOP * 6 * 4)
#define OFF_Y1     (OFF_X1 + KTOP * 4)
#define OFF_X2     (OFF_Y1 + KTOP * 4)
#define OFF_Y2     (OFF_X2 + KTOP * 4)
#define OFF_AREA   (OFF_Y2 + KTOP * 4)
#define OFF_VALIDW (OFF_AREA + KTOP * 4)
#define OFF_KEEPW  (OFF_VALIDW + WORDS * 4)
#define OFF_MASK   (OFF_KEEPW + WORDS * 4 + 48)  // 16B aligned; 2MB mask

__device__ __forceinline__ unsigned ballot32(bool p) {
#if __has_builtin(__builtin_amdgcn_ballot_w32)
  return __builtin_amdgcn_ballot_w32(p);
#else
  return (unsigned)__ballot(p);
#endif
}

// ---------------- k0: init counter + keys ----------------
__global__ void k_init(u32* __restrict__ cnt, u64* __restrict__ keys) {
  int t = blockIdx.x * blockDim.x + threadIdx.x;
  if (t == 0) *cnt = 0u;
  if (t < CAP) keys[t] = 0ull;
}

// ---------------- k1: sparse decode + append ----------------
// Reads only the obj-logit plane for all boxes (~1 MB of the 88 MB input);
// full 85-attr decode only for the ~1.4% that can pass sigmoid > 0.9
// (logit > ln 9 = 2.197 > 2.0 pre-test).
__global__ void k_decode(const float* __restrict__ x, u32* __restrict__ cnt,
                         u64* __restrict__ keys, float* __restrict__ slots) {
  int t = blockIdx.x * blockDim.x + threadIdx.x;
  if (t >= NBOX) return;
  int w = t % W_;
  int h = (t / W_) % H_;
  int a = (t / HW) % A_;
  int s = t / (A_ * HW);
  const float* base =
      x + ((size_t)s * (A_ * ATTRS) + (size_t)a * ATTRS) * HW + h * W_ + w;
  float tlogit = base[4 * HW];
  if (tlogit <= 2.0f) return;                      // conservative early-out
  float conf = 1.0f / (1.0f + expf(-tlogit));
  if (!(conf > OBJ_THR)) return;

  float tx = base[0 * HW], ty = base[1 * HW];
  float tw = base[2 * HW], th = base[3 * HW];
  const float aw[3] = {116.f / 8.f, 156.f / 8.f, 373.f / 8.f};
  const float ah[3] = {90.f / 8.f, 198.f / 8.f, 326.f / 8.f};
  float cx = (1.0f / (1.0f + expf(-tx)) + (float)w) * 8.0f;
  float cy = (1.0f / (1.0f + expf(-ty)) + (float)h) * 8.0f;
  float bw = aw[a] * expf(tw) * 8.0f;
  float bh = ah[a] * expf(th) * 8.0f;

  float best = base[5 * HW];
  int bi = 0;
#pragma unroll 4
  for (int c = 1; c < 80; ++c) {
    float v = base[(5 + c) * HW];
    if (v > best) { best = v; bi = c; }            // first-max like argmax
  }

  u32 slot = atomicAdd(cnt, 1u);
  if (slot >= CAP) return;
  float* sp = slots + (size_t)slot * 6;
  sp[0] = cx; sp[1] = cy; sp[2] = bw; sp[3] = bh; sp[4] = conf; sp[5] = (float)bi;
  // key: [62:31]=monotone(score)  [30:13]=inverted idx (tie -> low idx first)
  //      [12:0]=slot
  u32 mono = __float_as_uint(conf) | 0x80000000u;  // conf > 0
  u32 inv  = 262143u - (u32)t;                     // 18 bits, NBOX < 2^18
  keys[slot] = ((u64)mono << 31) | ((u64)inv << 13) | (u64)slot;
}

// ---------------- k2: bitonic sort 8192 keys (descending) ----------------
// Keys staged into LDS with CDNA5 async global->LDS DMA (ASYNCcnt path).
__global__ void __launch_bounds__(1024) k_sort(u64* __restrict__ keys) {
  __shared__ u64 sk[CAP];                // 64 KB, LDS offset 0 (only LDS obj)
  const int tid = threadIdx.x;
  {
    const char* g = (const char*)keys + (size_t)tid * 64;
    unsigned lbase = (unsigned)tid * 64u;
#pragma unroll
    for (int q = 0; q < 4; ++q) {
      unsigned lo = lbase + (unsigned)q * 16u;
      const void* gp = g + q * 16;
      asm volatile("global_load_async_to_lds_b128 %0, %1, off"
                   :: "v"(lo), "v"(gp) : "memory");
    }
    asm volatile("s_wait_asynccnt 0x0" ::: "memory");
  }
  __syncthreads();

  for (int k = 2; k <= CAP; k <<= 1) {
    for (int j = k >> 1; j > 0; j >>= 1) {
      for (int i = tid; i < CAP; i += 1024) {
        int ixj = i ^ j;
        if (ixj > i) {
          bool desc = ((i & k) == 0);
          u64 va = sk[i], vb = sk[ixj];
          bool sw = desc ? (va < vb) : (va > vb);
          if (sw) { sk[i] = vb; sk[ixj] = va; }
        }
      }
      __syncthreads();
    }
  }
  for (int i = tid; i < CAP; i += 1024) keys[i] = sk[i];
}

// ---------------- k3: gather top-K, corners/areas, valid bitmap ----------
__global__ void k_gather(const u64* __restrict__ keys,
                         const float* __restrict__ slots,
                         float* __restrict__ cand, float* __restrict__ x1,
                         float* __restrict__ y1, float* __restrict__ x2,
                         float* __restrict__ y2, float* __restrict__ area,
                         u32* __restrict__ validW) {
  int k = blockIdx.x * blockDim.x + threadIdx.x;   // KTOP threads
  u64 key = keys[k];
  u32 mono = (u32)(key >> 31);
  bool valid = mono > 0xBF666666u;                 // score > 0.9
  u32 slot = (u32)key & 0x1FFFu;
  float b0 = 0.f, b1 = 0.f, b2 = 0.f, b3 = 0.f, b4 = 0.f, b5 = 0.f;
  if (valid) {
    const float* sp = slots + (size_t)slot * 6;
    b0 = sp[0]; b1 = sp[1]; b2 = sp[2]; b3 = sp[3]; b4 = sp[4]; b5 = sp[5];
  }
  float* cp = cand + (size_t)k * 6;
  cp[0] = b0; cp[1] = b1; cp[2] = b2; cp[3] = b3; cp[4] = b4; cp[5] = b5;
  float hw = b2 * 0.5f, hh = b3 * 0.5f;
  x1[k] = b0 - hw; y1[k] = b1 - hh; x2[k] = b0 + hw; y2[k] = b1 + hh;
  area[k] = b2 * b3;
  unsigned bal = ballot32(valid);
  if ((threadIdx.x & 31) == 0) validW[k >> 5] = bal;
}

// ---------------- k4: suppression bitmask, WMMA area-sum tiles -----------
// One wave per 16(row) x 32(col) tile. Two V_WMMA_F32_16X16X4_F32 produce
// D[m][n] = area_row[m] + area_col[n] + 1e-9  (A=(a_i,1,..), B=(1,a_j,..)ᵀ,
// C = eps).  iou > 0.5  <=>  3*inter > D.  Rows beyond the valid count are
// never read by the scan, so those waves exit early (wave-uniform branch,
// EXEC all-ones at the WMMAs for surviving waves).
__global__ void __launch_bounds__(128) k_mask(
    const float* __restrict__ x1, const float* __restrict__ y1,
    const float* __restrict__ x2, const float* __restrict__ y2,
    const float* __restrict__ area, const u32* __restrict__ cnt,
    u32* __restrict__ mask) {
  const int lane = threadIdx.x & 31;
  const int wid = (blockIdx.x * 128 + threadIdx.x) >> 5;  // 0..32767
  const int tileR = wid >> 7;                             // 0..255
  const int tileC = wid & 127;                            // 0..127
  const int i0 = tileR * 16;
  const int j0 = tileC * 32;

  u32 vcnt = *cnt;                       // scalar (uniform) load
  int V = vcnt < KTOP ? (int)vcnt : KTOP;
  if (i0 >= V) return;                   // wave-uniform early exit

  const bool lowHalf = lane < 31 ? (lane < 16) : false;
  const int hl = lane & 15;

  // unconditional loads (always in range), select afterwards: no divergent
  // EXEC blocks before the WMMAs.
  float arRow = area[i0 + hl];
  float arColA = area[j0 + hl];
  float arColB = area[j0 + 16 + hl];

  // A 16x4: lanes0-15 K=0,1  lanes16-31 K=2,3 (zero)
  v2f afrag;
  afrag.x = lowHalf ? arRow : 0.0f;
  afrag.y = lowHalf ? 1.0f : 0.0f;
  // B 4x16: lanes0-15 K=0,1 rows  lanes16-31 K=2,3 (zero)
  v2f bfragA, bfragB;
  bfragA.x = lowHalf ? 1.0f : 0.0f;
  bfragA.y = lowHalf ? arColA : 0.0f;
  bfragB.x = bfragA.x;
  bfragB.y = lowHalf ? arColB : 0.0f;
  v8f c;
#pragma unroll
  for (int q = 0; q < 8; ++q) c[q] = 1e-9f;

  v8f dA = __builtin_amdgcn_wmma_f32_16x16x4_f32(false, afrag, false, bfragA,
                                                 (short)0, c, false, false);
  v8f dB = __builtin_amdgcn_wmma_f32_16x16x4_f32(false, afrag, false, bfragB,
                                                 (short)0, c, false, false);

  const int jA = j0 + hl, jB = jA + 16;
  float cx1A = x1[jA], cy1A = y1[jA], cx2A = x2[jA], cy2A = y2[jA];
  float cx1B = x1[jB], cy1B = y1[jB], cx2B = x2[jB], cy2B = y2[jB];
  const int rowOff = lowHalf ? 0 : 8;   // D: lanes16-31 hold M = v+8
  u32 myWord = 0u;
#pragma unroll
  for (int v = 0; v < 8; ++v) {
    int r = i0 + v + rowOff;
    float rx1 = x1[r], ry1 = y1[r], rx2 = x2[r], ry2 = y2[r];
    float ixA = fmaxf(fminf(rx2, cx2A) - fmaxf(rx1, cx1A), 0.f);
    float iyA = fmaxf(fminf(ry2, cy2A) - fmaxf(ry1, cy1A), 0.f);
    float ixB = fmaxf(fminf(rx2, cx2B) - fmaxf(rx1, cx1B), 0.f);
    float iyB = fmaxf(fminf(ry2, cy2B) - fmaxf(ry1, cy1B), 0.f);
    float interA = ixA * iyA, interB = ixB * iyB;
    bool supA = (3.0f * interA > dA[v]) && (jA > r);
    bool supB = (3.0f * interB > dB[v]) && (jB > r);
    u32 balA = ballot32(supA);
    u32 balB = ballot32(supB);
    u32 wLo = (balA & 0xFFFFu) | ((balB & 0xFFFFu) << 16);   // row i0+v
    u32 wHi = (balA >> 16) | (balB & 0xFFFF0000u);           // row i0+v+8
    if (lane == v) myWord = wLo;
    if (lane == v + 8) myWord = wHi;
  }
  if (lane < 16) mask[(size_t)(i0 + lane) * WORDS + tileC] = myWord;
}

// ---------------- k5: greedy scan, single wave, bitmap in VGPRs ----------
// Loop bounded by valid count: sorted order puts all valid candidates at
// the front, and keep[i] is false for i >= V, so trailing iterations are
// no-ops in the reference semantics.
__global__ void __launch_bounds__(32) k_scan(const u32* __restrict__ validW,
                                             const u32* __restrict__ mask,
                                             const u32* __restrict__ cnt,
                                             u32* __restrict__ keepW) {
  const int lane = threadIdx.x;                  // 0..31, 4 words each
  u32 vcnt = *cnt;
  const int V = vcnt < KTOP ? (int)vcnt : KTOP;
  uint4 vw = ((const uint4*)validW)[lane];
  u32 rem0 = ~vw.x, rem1 = ~vw.y, rem2 = ~vw.z, rem3 = ~vw.w;
  const uint4* mrows = (const uint4*)mask;       // 32 uint4 per 4096-bit row
  uint4 pre = mrows[lane];                       // prefetch row 0
  for (int i = 0; i < V; ++i) {
    uint4 m = pre;
    if (i + 1 < V) pre = mrows[(size_t)(i + 1) * 32 + lane];
    if (i + 2 < V)
      __builtin_prefetch(&mrows[(size_t)(i + 2) * 32 + lane], 0, 0);
    u32 q = (u32)(i >> 5) & 3u;
    u32 sel = (q == 0) ? rem0 : (q == 1) ? rem1 : (q == 2) ? rem2 : rem3;
    u32 own = (u32)__shfl((int)sel, i >> 7, 32);
    if (((own >> (i & 31)) & 1u) == 0u) {        // keep[i] -> suppress j>i
      rem0 |= m.x; rem1 |= m.y; rem2 |= m.z; rem3 |= m.w;
    }
  }
  uint4 kw;
  kw.x = ~rem0; kw.y = ~rem1; kw.z = ~rem2; kw.w = ~rem3;  // = valid & ~sup
  ((uint4*)keepW)[lane] = kw;
}

// ---------------- k6: emit padded output ----------------
__global__ void k_out(const float* __restrict__ cand,
                      const u32* __restrict__ keepW, float* __restrict__ out) {
  int t = blockIdx.x * blockDim.x + threadIdx.x;
  if (t >= KTOP * 6) return;
  int k = t / 6;
  bool kp = (keepW[k >> 5] >> (k & 31)) & 1u;
  out[t] = kp ? cand[t] : 0.0f;
}

// ---------------- launcher ----------------
extern "C" void kernel_launch(void* const* d_in, const int* in_sizes, int n_in,
                              void* d_out, int out_size, void* d_ws,
                              size_t ws_size, hipStream_t stream) {
  const float* x = (const float*)d_in[0];
  char* ws = (char*)d_ws;
  u32* cnt     = (u32*)(ws + OFF_CNT);
  u64* keys    = (u64*)(ws + OFF_KEYS);
  float* slots = (float*)(ws + OFF_SLOTS);
  float* cand  = (float*)(ws + OFF_CAND);
  float* X1    = (float*)(ws + OFF_X1);
  float* Y1    = (float*)(ws + OFF_Y1);
  float* X2    = (float*)(ws + OFF_X2);
  float* Y2    = (float*)(ws + OFF_Y2);
  float* AREA  = (float*)(ws + OFF_AREA);
  u32* validW  = (u32*)(ws + OFF_VALIDW);
  u32* keepW   = (u32*)(ws + OFF_KEEPW);
  u32* mask    = (u32*)(ws + OFF_MASK);
  float* out   = (float*)d_out;

  k_init<<<(CAP + 255) / 256, 256, 0, stream>>>(cnt, keys);
  k_decode<<<(NBOX + 255) / 256, 256, 0, stream>>>(x, cnt, keys, slots);
  k_sort<<<1, 1024, 0, stream>>>(keys);
  k_gather<<<KTOP / 128, 128, 0, stream>>>(keys, slots, cand, X1, Y1, X2, Y2,
                                           AREA, validW);
  k_mask<<<8192, 128, 0, stream>>>(X1, Y1, X2, Y2, AREA, cnt, mask);
  k_scan<<<1, 32, 0, stream>>>(validW, mask, cnt, keepW);
  k_out<<<(KTOP * 6 + 255) / 256, 256, 0, stream>>>(cand, keepW, out);
}